// CustomAttentionLayer_34282428956770
// MI455X (gfx1250) — compile-verified
//
#include <hip/hip_runtime.h>
#include <cstdint>

#define B_DIM 32
#define S_DIM 8192
#define D_DIM 256
#define K_TOP 1638            // max(1, int(8192 * 0.2))
#define EMPH  1.5f
#define NT    16              // S tiles in phase C
#define S_TILE (S_DIM / NT)   // 512

typedef float v8f __attribute__((ext_vector_type(8)));
typedef float v2f __attribute__((ext_vector_type(2)));

// Monotone map float -> uint (order-preserving), for exact k-th-largest select.
__device__ __forceinline__ unsigned f2key(float f) {
  unsigned u = __float_as_uint(f);
  return (u & 0x80000000u) ? ~u : (u | 0x80000000u);
}

// ---------------------------------------------------------------------------
// Phase A: e[bs] = tanh(x[bs,:] . W + bias) using V_WMMA_F32_16X16X4_F32.
// Framing: D = A x B, A(16x4) row 0 = W chunk, B(4x16)[k,n] = x[s_base+n, d0+k].
// Result row 0 (c[0], lanes 0..15) = raw dot products for 16 consecutive rows.
// One wave handles 16 rows; block = 8 waves = 128 rows.
// All masking is branchless (load unconditionally, v_cndmask the value) so the
// wave keeps EXEC full and the inner loop is a clean load+wmma stream.
// ---------------------------------------------------------------------------
__global__ __launch_bounds__(256) void phaseA_gemv_tanh(
    const float* __restrict__ x, const float* __restrict__ W,
    const float* __restrict__ bptr, float* __restrict__ e_buf) {
  const int tid  = threadIdx.x;
  const int lane = tid & 31;
  const int lm   = lane & 15;       // column index within tile
  const int half = lane >> 4;       // 0: K=0,1 ; 1: K=2,3
  const bool lm0 = (lm == 0);
  const long long bs_base = ((long long)blockIdx.x * 8 + (tid >> 5)) * 16;
  const float bias = bptr[0];

  // Per-lane base pointers (hoisted out of the loop).
  const float* xl = x + bs_base * D_DIM + (long long)lm * D_DIM + 2 * half;
  const float* wl = W + 2 * half;

  v8f c0 = {}; v8f c1 = {};
  for (int d0 = 0; d0 < D_DIM; d0 += 8) {
    // Unconditional loads (all addresses valid for every lane), then mask.
    const float w00 = wl[d0];
    const float w01 = wl[d0 + 1];
    const float w10 = wl[d0 + 4];
    const float w11 = wl[d0 + 5];
    const float2 xv0 = *(const float2*)(xl + d0);
    const float2 xv1 = *(const float2*)(xl + d0 + 4);

    v2f a0, a1, b0, b1;
    a0.x = lm0 ? w00 : 0.0f;  a0.y = lm0 ? w01 : 0.0f;
    a1.x = lm0 ? w10 : 0.0f;  a1.y = lm0 ? w11 : 0.0f;
    b0.x = xv0.x;             b0.y = xv0.y;
    b1.x = xv1.x;             b1.y = xv1.y;

    c0 = __builtin_amdgcn_wmma_f32_16x16x4_f32(false, a0, false, b0, (short)0, c0, false, false);
    c1 = __builtin_amdgcn_wmma_f32_16x16x4_f32(false, a1, false, b1, (short)0, c1, false, false);
  }
  const float e = tanhf(c0[0] + c1[0] + bias);
  if (lane < 16) e_buf[bs_base + lane] = e;
}

// ---------------------------------------------------------------------------
// Phase B: per-batch softmax stats + exact k-th-largest threshold on e
// (top-k of softmax(e) == top-k of e by monotonicity), then write
// emphasized_a straight into the output buffer. One 1024-thread block / batch.
// ---------------------------------------------------------------------------
__global__ __launch_bounds__(1024) void phaseB_softmax_topk(
    const float* __restrict__ e_buf, float* __restrict__ emph_out) {
  __shared__ float s_e[S_DIM];     // 32 KB of the 320 KB WGP pool
  __shared__ float s_red[32];
  __shared__ int   s_redi[32];
  __shared__ float s_bc;
  __shared__ int   s_cnt;
  const int tid  = threadIdx.x;
  const int lane = tid & 31;
  const int wid  = tid >> 5;
  const int b    = blockIdx.x;
  const float* eb = e_buf + (long long)b * S_DIM;

  for (int i = tid; i < S_DIM; i += 1024) s_e[i] = eb[i];
  __syncthreads();

  // --- max over S ---
  float m = -3.402823466e38f;
  for (int i = tid; i < S_DIM; i += 1024) m = fmaxf(m, s_e[i]);
  for (int off = 16; off; off >>= 1) m = fmaxf(m, __shfl_xor(m, off, 32));
  if (lane == 0) s_red[wid] = m;
  __syncthreads();
  if (tid == 0) {
    float t = s_red[0];
    for (int w = 1; w < 32; ++w) t = fmaxf(t, s_red[w]);
    s_bc = t;
  }
  __syncthreads();
  const float M = s_bc;
  __syncthreads();

  // --- sum of exp(e - M) ---
  float sm = 0.0f;
  for (int i = tid; i < S_DIM; i += 1024) sm += __expf(s_e[i] - M);
  for (int off = 16; off; off >>= 1) sm += __shfl_xor(sm, off, 32);
  if (lane == 0) s_red[wid] = sm;
  __syncthreads();
  if (tid == 0) {
    float t = 0.0f;
    for (int w = 0; w < 32; ++w) t += s_red[w];
    s_bc = t;
  }
  __syncthreads();
  const float inv_denom = 1.0f / s_bc;
  __syncthreads();

  // --- k-th largest key via 32-step radix binary search (exact select) ---
  unsigned v = 0;
  for (int bit = 31; bit >= 0; --bit) {
    const unsigned trial = v | (1u << bit);
    int cnt = 0;
    for (int i = tid; i < S_DIM; i += 1024) cnt += (f2key(s_e[i]) >= trial) ? 1 : 0;
    for (int off = 16; off; off >>= 1) cnt += __shfl_xor(cnt, off, 32);
    if (lane == 0) s_redi[wid] = cnt;
    __syncthreads();
    if (tid == 0) {
      int t = 0;
      for (int w = 0; w < 32; ++w) t += s_redi[w];
      s_cnt = t;
    }
    __syncthreads();
    if (s_cnt >= K_TOP) v = trial;   // uniform across block
    __syncthreads();
  }

  // --- emphasized softmax out ---
  float* out = emph_out + (long long)b * S_DIM;
  for (int i = tid; i < S_DIM; i += 1024) {
    const float e = s_e[i];
    const float a = __expf(e - M) * inv_denom;
    out[i] = (f2key(e) >= v) ? a * EMPH : a;
  }
}

// ---------------------------------------------------------------------------
// Phase C: partial[b,t,d] = sum_{s in tile t} emph_a[b,s] * x[b,s,d] via WMMA.
// A(16x4) row 0 = emph_a chunk (from LDS), B(4x16)[k,n] = x[s0+k, d0+n]
// (coalesced 64B rows). 16 waves/block cover all 256 d. 512 blocks total.
// Branchless masking as in phase A.
// ---------------------------------------------------------------------------
__global__ __launch_bounds__(512) void phaseC_weighted_sum(
    const float* __restrict__ x, const float* __restrict__ emph,
    float* __restrict__ partial) {
  __shared__ float s_a[S_TILE];
  const int tid  = threadIdx.x;
  const int lane = tid & 31;
  const int lm   = lane & 15;
  const int half = lane >> 4;
  const bool lm0 = (lm == 0);
  const int wave = tid >> 5;              // 0..15 -> d tile
  const int b    = blockIdx.x / NT;
  const int t    = blockIdx.x % NT;
  const int s0   = t * S_TILE;

  for (int i = tid; i < S_TILE; i += 512)
    s_a[i] = emph[(long long)b * S_DIM + s0 + i];
  __syncthreads();

  const int d0 = wave * 16;
  // Per-lane base pointer: row (s + 2*half), column (d0 + lm).
  const float* xl = x + ((long long)b * S_DIM + s0 + 2 * half) * D_DIM + d0 + lm;
  const float* al = s_a + 2 * half;

  v8f c0 = {}; v8f c1 = {};
  for (int s = 0; s < S_TILE; s += 8) {
    // Unconditional LDS reads, masked by value.
    const float sa00 = al[s];
    const float sa01 = al[s + 1];
    const float sa10 = al[s + 4];
    const float sa11 = al[s + 5];
    // Coalesced global loads: 16 consecutive floats per half-wave.
    const long long r = (long long)s * D_DIM;
    const float xv00 = xl[r];
    const float xv01 = xl[r + D_DIM];
    const float xv10 = xl[r + 4 * D_DIM];
    const float xv11 = xl[r + 5 * D_DIM];

    v2f a0, a1, b0, b1;
    a0.x = lm0 ? sa00 : 0.0f;  a0.y = lm0 ? sa01 : 0.0f;
    a1.x = lm0 ? sa10 : 0.0f;  a1.y = lm0 ? sa11 : 0.0f;
    b0.x = xv00;               b0.y = xv01;
    b1.x = xv10;               b1.y = xv11;

    c0 = __builtin_amdgcn_wmma_f32_16x16x4_f32(false, a0, false, b0, (short)0, c0, false, false);
    c1 = __builtin_amdgcn_wmma_f32_16x16x4_f32(false, a1, false, b1, (short)0, c1, false, false);
  }
  const float r = c0[0] + c1[0];
  if (lane < 16)
    partial[((long long)(b * NT + t)) * D_DIM + d0 + lane] = r;
}

// ---------------------------------------------------------------------------
// Final deterministic reduction of the NT partials per (b,d).
// ---------------------------------------------------------------------------
__global__ __launch_bounds__(256) void reduce_partials(
    const float* __restrict__ partial, float* __restrict__ out) {
  const int idx = blockIdx.x * 256 + threadIdx.x;   // 0 .. B*D-1
  const int b = idx / D_DIM;
  const int d = idx % D_DIM;
  float s = 0.0f;
  for (int t = 0; t < NT; ++t)
    s += partial[((long long)(b * NT + t)) * D_DIM + d];
  out[idx] = s;
}

extern "C" void kernel_launch(void* const* d_in, const int* in_sizes, int n_in,
                              void* d_out, int out_size, void* d_ws, size_t ws_size,
                              hipStream_t stream) {
  (void)in_sizes; (void)n_in; (void)out_size; (void)ws_size;
  const float* x    = (const float*)d_in[0];   // [B,S,D] fp32
  const float* W    = (const float*)d_in[1];   // [D,1]   fp32
  const float* bias = (const float*)d_in[2];   // [1]     fp32
  float* out  = (float*)d_out;                 // [B*D] summed ++ [B*S] emph_a
  float* emph = out + (long long)B_DIM * D_DIM;

  float* ws      = (float*)d_ws;
  float* e_buf   = ws;                                  // B*S floats
  float* partial = ws + (long long)B_DIM * S_DIM;       // B*NT*D floats

  phaseA_gemv_tanh<<<(B_DIM * S_DIM) / 128, 256, 0, stream>>>(x, W, bias, e_buf);
  phaseB_softmax_topk<<<B_DIM, 1024, 0, stream>>>(e_buf, emph);
  phaseC_weighted_sum<<<B_DIM * NT, 512, 0, stream>>>(x, emph, partial);
  reduce_partials<<<(B_DIM * D_DIM) / 256, 256, 0, stream>>>(partial, out);
}